// FourierLayer_70463233458452
// MI455X (gfx1250) — compile-verified
//
#include <hip/hip_runtime.h>

// ---------------------------------------------------------------------------
// FNO spectral layer as a chain of WMMA f32 GEMMs (gfx1250 / CDNA5, wave32).
//   step1: truncated forward DFT over W axis   (K=512, x staged in LDS, f4 loads)
//   step2: truncated forward DFT over H axis   (K=1024, complex folded into K)
//   step3: per-mode CxC mixing with real W     (K=32)
//   step4: zero-padded inverse DFT over H axis (K=64, B staged via TDM)
//   step5: zero-padded inverse rDFT over W axis(K=64, B staged via TDM)
// ---------------------------------------------------------------------------

typedef __attribute__((ext_vector_type(2))) float v2f;
typedef __attribute__((ext_vector_type(8))) float v8f;
typedef __attribute__((ext_vector_type(4))) unsigned int u32x4;
typedef __attribute__((ext_vector_type(4))) int i32x4;
typedef __attribute__((ext_vector_type(8))) int i32x8;
typedef __attribute__((address_space(3))) float lds_float;

#define PI2 6.28318530717958647692f

// workspace layout (float offsets)
constexpr size_t OFF_T1RE = 0;                       // 32*512  E.re  (step1)
constexpr size_t OFF_T1IM = OFF_T1RE + 32 * 512;     // 32*512  E.im
constexpr size_t OFF_A2RE = OFF_T1IM + 32 * 512;     // 32*1024 [cos| sin]
constexpr size_t OFF_A2IM = OFF_A2RE + 32 * 1024;    // 32*1024 [-sin|cos]
constexpr size_t OFF_A4RE = OFF_A2IM + 32 * 1024;    // 512*64  [cos|-sin]/512
constexpr size_t OFF_A4IM = OFF_A4RE + 512 * 64;     // 512*64  [sin| cos]/512
constexpr size_t OFF_A5   = OFF_A4IM + 512 * 64;     // 512*64  c_k*[cos|-sin]/512
constexpr size_t OFF_G    = OFF_A5   + 512 * 64;     // 8*2*512*32*32
constexpr size_t OFF_LOW  = OFF_G    + 8ull * 2 * 512 * 32 * 32;  // 8*2*32*1024
constexpr size_t OFF_MIX  = OFF_LOW  + 8ull * 2 * 32 * 1024;      // 8*32*2*32*32
constexpr size_t OFF_T    = OFF_MIX  + 8ull * 32 * 2 * 32 * 32;   // 8*512*2*32*32

// ---- WMMA helpers ---------------------------------------------------------

__device__ __forceinline__ v8f wmma4(v2f a, v2f b, v8f c) {
  // D(16x16,f32) = A(16x4,f32) * B(4x16,f32) + C
  return __builtin_amdgcn_wmma_f32_16x16x4_f32(false, a, false, b,
                                               (short)0, c, false, false);
}

// C(16x16) += A(16xK, row-major lda) * B(Kx16, row-major ldb)  [global B]
__device__ __forceinline__ v8f wmma_gemm(const float* __restrict__ A, int lda,
                                         const float* __restrict__ B, int ldb,
                                         int K, v8f acc) {
  const int lane = threadIdx.x & 31;
  const int half = lane >> 4;   // 0: K sub 0,1   1: K sub 2,3
  const int idx  = lane & 15;   // A row / B col
  const float* ap = A + (size_t)idx * lda + half * 2;
  const float* bp = B + (size_t)(half * 2) * ldb + idx;
#pragma unroll 4
  for (int k0 = 0; k0 < K; k0 += 4) {
    v2f a, b;
    a = *(const v2f*)ap;          // A[row, k0+2h], A[row, k0+2h+1]
    b.x = bp[0];                  // B[k0+2h,   col]
    b.y = bp[ldb];                // B[k0+2h+1, col]
    acc = wmma4(a, b, acc);
    ap += 4;
    bp += 4 * (size_t)ldb;
  }
  return acc;
}

// store 16x16 f32 tile: VGPR v <-> row (v + 8*half), col idx
__device__ __forceinline__ void store_tile(float* __restrict__ C, int ldc, v8f acc) {
  const int lane = threadIdx.x & 31;
  const int half = lane >> 4;
  const int idx  = lane & 15;
#pragma unroll
  for (int v = 0; v < 8; ++v)
    C[(size_t)(v + half * 8) * ldc + idx] = acc[v];
}

// ---- TDM: async-copy 2048 contiguous floats (8KB) from global into LDS ----
__device__ __forceinline__ void tdm_load_2048f(unsigned lds_off,
                                               const float* __restrict__ src) {
  unsigned long long ga = (unsigned long long)(size_t)src;
  u32x4 g0;
  g0[0] = 1u;                                     // count=1, user mode
  g0[1] = lds_off;                                // lds_addr (bytes)
  g0[2] = (unsigned)(ga & 0xffffffffu);           // global_addr[31:0]
  g0[3] = (unsigned)((ga >> 32) & 0x01ffffffu)    // global_addr[56:32]
          | 0x80000000u;                          // type = 2 ("image")
  i32x8 g1 = {0, 0, 0, 0, 0, 0, 0, 0};
  g1[0] = 0x20000;                                // data_size = 2 (4 bytes)
  g1[1] = (int)(2048u << 16);                     // tensor_dim0 = 2048 (lo16)
  g1[2] = (int)(1u << 16);                        // tensor_dim1 = 1
  g1[3] = (int)(2048u << 16);                     // tile_dim0 = 2048
  g1[4] = 1;                                      // tile_dim1 = 1
  g1[5] = 2048;                                   // tensor_dim0_stride
  i32x4 z4 = {0, 0, 0, 0};
#if defined(__clang_major__) && __clang_major__ >= 23
  i32x8 z8 = {0, 0, 0, 0, 0, 0, 0, 0};
  __builtin_amdgcn_tensor_load_to_lds(g0, g1, z4, z4, z8, 0);
#else
  __builtin_amdgcn_tensor_load_to_lds(g0, g1, z4, z4, 0);
#endif
}

__device__ __forceinline__ unsigned lds_offset_of(const float* p) {
  // addrspacecast generic->LDS, then ptr->u64->u32 (AS(3) ptr is 32-bit on
  // device; the u64 hop keeps the host-side parse of this function legal).
  return (unsigned)(unsigned long long)(lds_float*)p;
}

// ---- table construction ---------------------------------------------------

__global__ void fno_tables(float* __restrict__ ws) {
  const int t = blockIdx.x * 256 + threadIdx.x;
  const float base = PI2 / 512.0f;
  if (t < 32768) {                          // T1 re/im: [k][w], e^{-i 2pi k w/512}
    int im = t >> 14, r = t & 16383;
    int k = r >> 9, w = r & 511;
    float th = base * (float)((k * w) & 511);
    ws[OFF_T1RE + t] = im ? -__sinf(th) : __cosf(th);
  } else if (t < 98304) {                   // A2 re/im: [kx][part*512+h]
    int u = t - 32768;
    int im = u >> 15, r = u & 32767;
    int kx = r >> 10, col = r & 1023;
    int part = col >> 9, h = col & 511;
    float th = base * (float)((kx * h) & 511);
    float c = __cosf(th), s = __sinf(th);
    ws[OFF_A2RE + u] = im ? (part ? c : -s) : (part ? s : c);
  } else if (t < 163840) {                  // A4 re/im: [h][part*32+kx], /512
    int u = t - 98304;
    int im = u >> 15, r = u & 32767;
    int h = r >> 6, col = r & 63;
    int part = col >> 5, kx = col & 31;
    float th = base * (float)((h * kx) & 511);
    float c = __cosf(th) * (1.0f / 512.0f);
    float s = __sinf(th) * (1.0f / 512.0f);
    ws[OFF_A4RE + u] = im ? (part ? c : s) : (part ? -s : c);
  } else if (t < 196608) {                  // A5: [w][part*32+ky], c_k weights, /512
    int u = t - 163840;
    int w = u >> 6, col = u & 63;
    int part = col >> 5, ky = col & 31;
    float sc = (ky == 0 ? 1.0f : 2.0f) * (1.0f / 512.0f);
    float th = base * (float)((ky * w) & 511);
    ws[OFF_A5 + u] = part ? -sc * __sinf(th) : sc * __cosf(th);
  }
}

// ---- step 1: G[b,part,h,ky,c] = sum_w E[ky,w] * x[b,h,w,c] ---------------
// One block per (b,h). x row (512x32) staged into LDS transposed [c][w] in
// 64-row double-buffered chunks (float4 reads); 8 waves = {part}x{ky}x{c} tiles.
__global__ void __launch_bounds__(256)
fno_step1(const float* __restrict__ x, float* __restrict__ ws) {
  __shared__ float sX[2][32 * 66];
  const int bh = blockIdx.x;                // b*512 + h
  const int b = bh >> 9, h = bh & 511;
  const int wid = threadIdx.x >> 5;
  const int part = wid & 1, ti = (wid >> 1) & 1, tj = wid >> 2;
  const int lane = threadIdx.x & 31, half = lane >> 4, idx = lane & 15;
  const float* Arow = ws + (part ? OFF_T1IM : OFF_T1RE) +
                      (size_t)(ti * 16 + idx) * 512 + half * 2;
  const float* xrow = x + (size_t)bh * (512 * 32);
  const int bbase = (tj * 16 + idx) * 66 + half * 2;

  for (int i = threadIdx.x; i < 512; i += 256) {    // stage chunk 0 (f4)
    float4 v = ((const float4*)xrow)[i];
    int wl = i >> 3, c = (i & 7) * 4;
    float* d0 = &sX[0][c * 66 + wl];
    d0[0] = v.x; d0[66] = v.y; d0[132] = v.z; d0[198] = v.w;
  }
  __syncthreads();

  v8f acc = {};
#pragma unroll 1
  for (int chunk = 0; chunk < 8; ++chunk) {
    if (chunk < 7) {                                // prefetch next chunk
      const float4* src = (const float4*)(xrow + (chunk + 1) * 2048);
      float* dstb = sX[(chunk + 1) & 1];
      for (int i = threadIdx.x; i < 512; i += 256) {
        float4 v = src[i];
        int wl = i >> 3, c = (i & 7) * 4;
        float* d0 = &dstb[c * 66 + wl];
        d0[0] = v.x; d0[66] = v.y; d0[132] = v.z; d0[198] = v.w;
      }
    }
    const float* sxc = sX[chunk & 1];
    const float* ap = Arow + chunk * 64;
#pragma unroll
    for (int k0 = 0; k0 < 64; k0 += 4) {
      v2f a = *(const v2f*)(ap + k0);
      v2f bb;
      bb.x = sxc[bbase + k0];
      bb.y = sxc[bbase + k0 + 1];
      acc = wmma4(a, bb, acc);
    }
    __syncthreads();
  }
  float* C = ws + OFF_G +
             ((((size_t)b * 2 + part) * 512 + h) * 32 + ti * 16) * 32 + tj * 16;
  store_tile(C, 32, acc);
}

// ---- step 2: low[b,part,kx,(ky,c)] = A2part(32x1024) x G_b(1024x1024) ----
__global__ void __launch_bounds__(256) fno_step2(float* __restrict__ ws) {
  const int q = blockIdx.x * 8 + (threadIdx.x >> 5);   // 0..2047
  const int b = q >> 8;
  int r = q & 255;
  const int part = r >> 7; r &= 127;
  const int ti = r >> 6, tj = r & 63;
  const float* A  = ws + (part ? OFF_A2IM : OFF_A2RE) + (size_t)ti * 16 * 1024;
  const float* Bm = ws + OFF_G + (size_t)b * (2 * 512 * 1024) + tj * 16;
  v8f acc = {};
  acc = wmma_gemm(A, 1024, Bm, 1024, 1024, acc);
  float* C = ws + OFF_LOW + ((size_t)(b * 2 + part) * 32 + ti * 16) * 1024 + tj * 16;
  store_tile(C, 1024, acc);
}

// ---- step 3: mixed[(part,b),d] = low[(part,b),c] x W[kx,ky](32x32) -------
__global__ void __launch_bounds__(256)
fno_step3(const float* __restrict__ Wt, float* __restrict__ ws) {
  const int q = blockIdx.x * 8 + (threadIdx.x >> 5);   // 0..2047
  const int tj = q & 1;
  const int kxky = q >> 1;
  const int kx = kxky >> 5, ky = kxky & 31;
  const int lane = threadIdx.x & 31;
  const int half = lane >> 4, idx = lane & 15;
  const int partA = idx >> 3, bA = idx & 7;            // row m = part*8 + b
  const float* Arow = ws + OFF_LOW +
                      ((size_t)(bA * 2 + partA) * 32 + kx) * 1024 + ky * 32;
  const float* Bm = Wt + ((size_t)kx * 32 + ky) * 1024 + tj * 16;
  v8f acc = {};
#pragma unroll
  for (int k0 = 0; k0 < 32; k0 += 4) {
    v2f a, bb;
    a = *(const v2f*)(Arow + k0 + half * 2);
    const float* bp = Bm + (size_t)(k0 + half * 2) * 32 + idx;
    bb.x = bp[0];
    bb.y = bp[32];
    acc = wmma4(a, bb, acc);
  }
#pragma unroll
  for (int v = 0; v < 8; ++v) {             // mixed[b][ky][part][kx][d]
    int m = v + half * 8, part = m >> 3, b = m & 7;
    ws[OFF_MIX + (((size_t)(b * 32 + ky) * 2 + part) * 32 + kx) * 32 + tj * 16 + idx] =
        acc[v];
  }
}

// ---- step 4: T.part[h,d](512x32) = A4.part(512x64) x mixed_{b,ky}(64x32) --
// One block per (b,ky); B (8KB, [k][d]) staged via TDM, held in registers per
// wave (16 v2f); waves = {tj} x {part} x {th group}.
__global__ void __launch_bounds__(256) fno_step4(float* __restrict__ ws) {
  __shared__ float sB[2048];
  const int bky = blockIdx.x;               // b*32 + ky
  const int b = bky >> 5, ky = bky & 31;
  const int wid = threadIdx.x >> 5;
  const int lane = threadIdx.x & 31, half = lane >> 4, idx = lane & 15;
  if (threadIdx.x < 32) {                   // wave 0: issue TDM, await done
    tdm_load_2048f(lds_offset_of(sB), ws + OFF_MIX + (size_t)bky * 2048);
    __builtin_amdgcn_s_wait_tensorcnt(0);
  }
  __syncthreads();
  const int tj = wid & 1;
  const int part = (wid >> 1) & 1;
  const int thg = wid >> 2;                 // 0..1
  v2f bf[16];
  const int col = tj * 16 + idx;
#pragma unroll
  for (int k0 = 0; k0 < 16; ++k0) {        // B[k][d] -> per-wave registers
    bf[k0].x = sB[(k0 * 4 + half * 2) * 32 + col];
    bf[k0].y = sB[(k0 * 4 + half * 2 + 1) * 32 + col];
  }
  const float* Atab = ws + (part ? OFF_A4IM : OFF_A4RE);
#pragma unroll 1
  for (int j = 0; j < 16; ++j) {
    const int th = thg * 16 + j;
    const float* ap = Atab + (size_t)(th * 16 + idx) * 64 + half * 2;
    v8f acc = {};
#pragma unroll
    for (int k0 = 0; k0 < 16; ++k0)
      acc = wmma4(*(const v2f*)(ap + k0 * 4), bf[k0], acc);
    float* C = ws + OFF_T +
               (((size_t)(b * 512 + th * 16) * 2 + part) * 32 + ky) * 32 + tj * 16;
    store_tile(C, 2048, acc);               // h stride = 2*32*32
  }
}

// ---- step 5: out[w,d](512x32) = A5(512x64) x T_{b,h}(64x32) ---------------
// One block per (b,h); B (8KB) staged via TDM, register-cached per wave;
// each wave sweeps 8 w-tiles against its fixed B fragments.
__global__ void __launch_bounds__(256)
fno_step5(float* __restrict__ out, const float* __restrict__ ws) {
  __shared__ float sB[2048];
  const int bh = blockIdx.x;                // b*512 + h
  const int wid = threadIdx.x >> 5;
  const int lane = threadIdx.x & 31, half = lane >> 4, idx = lane & 15;
  if (threadIdx.x < 32) {                   // wave 0: issue TDM, await done
    tdm_load_2048f(lds_offset_of(sB), ws + OFF_T + (size_t)bh * 2048);
    __builtin_amdgcn_s_wait_tensorcnt(0);
  }
  __syncthreads();
  const int tj = wid & 1;
  const int twg = wid >> 1;                 // 0..3
  v2f bf[16];
  const int col = tj * 16 + idx;
#pragma unroll
  for (int k0 = 0; k0 < 16; ++k0) {        // B[k][d] -> per-wave registers
    bf[k0].x = sB[(k0 * 4 + half * 2) * 32 + col];
    bf[k0].y = sB[(k0 * 4 + half * 2 + 1) * 32 + col];
  }
  float* Cbh = out + (size_t)bh * (512 * 32) + tj * 16;
#pragma unroll 1
  for (int j = 0; j < 8; ++j) {
    const int tw = twg * 8 + j;
    const float* ap = ws + OFF_A5 + (size_t)(tw * 16 + idx) * 64 + half * 2;
    v8f acc = {};
#pragma unroll
    for (int k0 = 0; k0 < 16; ++k0)
      acc = wmma4(*(const v2f*)(ap + k0 * 4), bf[k0], acc);
    store_tile(Cbh + (size_t)tw * 16 * 32, 32, acc);
  }
}

extern "C" void kernel_launch(void* const* d_in, const int* in_sizes, int n_in,
                              void* d_out, int out_size, void* d_ws, size_t ws_size,
                              hipStream_t stream) {
  const float* x = (const float*)d_in[0];   // (8,512,512,32) f32
  const float* W = (const float*)d_in[1];   // (32,32,32,32)  f32
  float* out = (float*)d_out;               // (8,512,512,32) f32
  float* ws = (float*)d_ws;                 // needs ~72 MB

  fno_tables<<<768, 256, 0, stream>>>(ws);
  fno_step1<<<4096, 256, 0, stream>>>(x, ws);
  fno_step2<<<256, 256, 0, stream>>>(ws);
  fno_step3<<<256, 256, 0, stream>>>(W, ws);
  fno_step4<<<256, 256, 0, stream>>>(ws);
  fno_step5<<<4096, 256, 0, stream>>>(out, ws);
}